// SelfAttentionLayer_35897336660592
// MI455X (gfx1250) — compile-verified
//
#include <hip/hip_runtime.h>

// out = X @ Wv^T + bv   (softmax row-sums are identically 1, so attention collapses to V)
// M = 16384, N = K = 1024, fp32 throughout via V_WMMA_F32_16X16X4_F32.

typedef float v2f __attribute__((ext_vector_type(2)));
typedef float v8f __attribute__((ext_vector_type(8)));

#define KDIM 1024
#define NDIM 1024
#define BM 128
#define BN 128
#define BK 32
#define LDSS 36   // padded row stride (dwords): 144B rows -> 16B aligned, conflict-free frag reads

__global__ __launch_bounds__(256)
void attn_v_gemm_f32_wmma(const float* __restrict__ X,
                          const float* __restrict__ Wv,
                          const float* __restrict__ bv,
                          float* __restrict__ out)
{
    __shared__ float sA[BM * LDSS];   // X tile:  128 rows x 32 k
    __shared__ float sB[BN * LDSS];   // Wv tile: 128 rows x 32 k (Wv is (N,K) row-major)

    const int tid     = threadIdx.x;
    const int lane    = tid & 31;
    const int wave    = tid >> 5;     // 0..7
    const int waveRow = wave >> 2;    // 0..1 -> 64-row strip of the block tile
    const int waveCol = wave & 3;     // 0..3 -> 32-col strip

    const int mBlock = blockIdx.y * BM;
    const int nBlock = blockIdx.x * BN;

    // global->LDS mapping: 256 threads, float4 per load, 8 threads cover a 32-float row
    const int ldRow = tid >> 3;          // 0..31
    const int ldCol = (tid & 7) * 4;     // 0,4,...,28

    v8f acc[4][2];
    const v8f vzero = {0.f, 0.f, 0.f, 0.f, 0.f, 0.f, 0.f, 0.f};
#pragma unroll
    for (int i = 0; i < 4; ++i)
#pragma unroll
        for (int j = 0; j < 2; ++j)
            acc[i][j] = vzero;

    const int halfK = 2 * (lane >> 4);   // lanes 16..31 handle K+2,K+3 of each 4-wide step
    const int subM  = lane & 15;

    for (int kb = 0; kb < KDIM; kb += BK) {
        // ---- stage tiles into LDS ----
#pragma unroll
        for (int p = 0; p < 4; ++p) {
            const int r = p * 32 + ldRow;
            const float4 a = *(const float4*)(X  + (size_t)(mBlock + r) * KDIM + kb + ldCol);
            const float4 b = *(const float4*)(Wv + (size_t)(nBlock + r) * KDIM + kb + ldCol);
            *(float4*)&sA[r * LDSS + ldCol] = a;
            *(float4*)&sB[r * LDSS + ldCol] = b;
        }
        __syncthreads();

        // ---- 8 WMMA k-steps of 4 over this 32-wide chunk ----
#pragma unroll
        for (int ks = 0; ks < 8; ++ks) {
            const int kc = ks * 4 + halfK;        // even -> 8B-aligned ds_load_b64
            v2f aF[4], bF[2];
            const int rBase = waveRow * 64 + subM;
#pragma unroll
            for (int i = 0; i < 4; ++i)
                aF[i] = *(const v2f*)&sA[(rBase + i * 16) * LDSS + kc];
            const int nBase = waveCol * 32 + subM;
#pragma unroll
            for (int j = 0; j < 2; ++j)
                bF[j] = *(const v2f*)&sB[(nBase + j * 16) * LDSS + kc];

#pragma unroll
            for (int i = 0; i < 4; ++i)
#pragma unroll
                for (int j = 0; j < 2; ++j)
                    acc[i][j] = __builtin_amdgcn_wmma_f32_16x16x4_f32(
                        /*neg_a=*/false, aF[i], /*neg_b=*/false, bF[j],
                        /*c_mod=*/(short)0, acc[i][j],
                        /*reuse_a=*/false, /*reuse_b=*/false);
        }
        __syncthreads();
    }

    // ---- epilogue: bias add + store (C/D layout: vgpr p, lane L -> m = p + 8*(L>>4), n = L&15) ----
    const int mHalf = 8 * (lane >> 4);
#pragma unroll
    for (int i = 0; i < 4; ++i) {
        const int rowTile = mBlock + waveRow * 64 + i * 16 + mHalf;
#pragma unroll
        for (int j = 0; j < 2; ++j) {
            const int n = nBlock + waveCol * 32 + j * 16 + subM;
            const float bias = bv[n];
#pragma unroll
            for (int p = 0; p < 8; ++p)
                out[(size_t)(rowTile + p) * NDIM + n] = acc[i][j][p] + bias;
        }
    }
}

extern "C" void kernel_launch(void* const* d_in, const int* in_sizes, int n_in,
                              void* d_out, int out_size, void* d_ws, size_t ws_size,
                              hipStream_t stream) {
    // inputs: X, Wq, bq, Wk, bk, Wv, bv  (only X, Wv, bv matter: out == V exactly)
    const float* X  = (const float*)d_in[0];
    const float* Wv = (const float*)d_in[5];
    const float* bv = (const float*)d_in[6];
    float* out = (float*)d_out;

    const int M = in_sizes[0] / KDIM;   // 8 * 2048 = 16384
    dim3 grid(NDIM / BN, M / BM);       // 8 x 128 = 1024 blocks
    dim3 block(256);                    // 8 wave32 waves
    hipLaunchKernelGGL(attn_v_gemm_f32_wmma, grid, block, 0, stream, X, Wv, bv, out);
}